// DelayedXOR_SH_SNN_Legacy_2637109920463
// MI455X (gfx1250) — compile-verified
//
#include <hip/hip_runtime.h>

// SNN LIF 2-layer scan on gfx1250.
// One wave32 = one 16-batch tile. Layer-1 GEMM via V_WMMA_F32_16X16X4_F32:
//   A (16x4, M=hidden, K=input padded 2->4) = 0.5*W1, constant over t
//   B (4x16,  K=input, N=batch)             = x[b, t, :]
//   C (16x16)                               = 0.5*b1 broadcast (bias for free)
//   D[M=hidden, N=batch]                    = 0.5*h1
// Membrane update v = 0.5*v + 0.5*h is then one FMA per accumulator VGPR.
// Layer-2 (H=32 -> O=1) is an in-lane sum of spike-selected 0.5*W2 weights
// plus a single cross-half shfl_xor(16), since each lane holds 16 of the 32
// hidden rows for its batch column and its half-partner lane holds the rest.

typedef __attribute__((ext_vector_type(2))) float v2f;
typedef __attribute__((ext_vector_type(8))) float v8f;

__global__ __launch_bounds__(256) void snn_lif_wmma_kernel(
    const float* __restrict__ x,    // [B, T, 2]
    const float* __restrict__ W1,   // [32, 2]
    const float* __restrict__ b1,   // [32]
    const float* __restrict__ W2,   // [1, 32]
    const float* __restrict__ b2,   // [1]
    float* __restrict__ out,        // [B, 1]
    int T, int dstart, int numTiles)
{
  const int lane = threadIdx.x & 31;
  const int wave = threadIdx.x >> 5;
  const int tile = blockIdx.x * (blockDim.x >> 5) + wave;
  if (tile >= numTiles) return;          // wave-uniform: EXEC stays all-ones

  const int m    = lane & 15;            // A: hidden row / D: batch column
  const int half = lane >> 4;            // 0: K=0,1  1: K=2,3 (pad lanes)
  const int b0   = tile * 16;

  // ---- constant WMMA operands (1/TAU = 0.5 folded into all weights) ------
  const float hm = half ? 0.0f : 0.5f;   // zero the K=2,3 pad lanes of A
  v2f aW0, aW1;                          // tile0: hidden 0..15, tile1: 16..31
  aW0.x = hm * W1[m * 2 + 0];
  aW0.y = hm * W1[m * 2 + 1];
  aW1.x = hm * W1[(16 + m) * 2 + 0];
  aW1.y = hm * W1[(16 + m) * 2 + 1];

  v8f c0, c1;                            // C = 0.5*b1 : bias added by WMMA
  float w20[8], w21[8];                  // 0.5*W2 for the hidden rows we own
  #pragma unroll
  for (int r = 0; r < 8; ++r) {
    const int h = r + 8 * half;          // D row M owned by this lane at VGPR r
    c0[r]  = 0.5f * b1[h];
    c1[r]  = 0.5f * b1[16 + h];
    w20[r] = 0.5f * W2[h];
    w21[r] = 0.5f * W2[16 + h];
  }
  const float b2h = 0.5f * b2[0];

  // ---- recurrent state (all in VGPRs) ------------------------------------
  float v1a[8], v1b[8];
  #pragma unroll
  for (int r = 0; r < 8; ++r) { v1a[r] = 0.0f; v1b[r] = 0.0f; }
  float v2m = 0.0f;                      // layer-2 membrane (batch = lane&15)
  float acc = 0.0f;                      // spike count in decision window

  const float* xb = x + (size_t)(b0 + m) * (size_t)T * 2u;

  auto lif_step = [&](float x0, float x1, bool window) {
    v2f bx; bx.x = x0; bx.y = x1;        // pad lanes' garbage * A-zeros = 0
    v8f h0 = __builtin_amdgcn_wmma_f32_16x16x4_f32(
        false, aW0, false, bx, (short)0, c0, false, false);
    v8f h1 = __builtin_amdgcn_wmma_f32_16x16x4_f32(
        false, aW1, false, bx, (short)0, c1, false, false);

    float hsum = 0.0f;                   // sum_h s1[h] * 0.5*W2[h] (partial)
    #pragma unroll
    for (int r = 0; r < 8; ++r) {
      float va = fmaf(0.5f, v1a[r], h0[r]);   // v1 = 0.5 v1 + 0.5 h1
      bool  sa = va >= 1.0f;                  // Heaviside(v1 - VTH)
      hsum += sa ? w20[r] : 0.0f;
      v1a[r] = sa ? 0.0f : va;                // hard reset
      float vb = fmaf(0.5f, v1b[r], h1[r]);
      bool  sb = vb >= 1.0f;
      hsum += sb ? w21[r] : 0.0f;
      v1b[r] = sb ? 0.0f : vb;
    }
    hsum += __shfl_xor(hsum, 16, 32);    // combine the two hidden halves

    v2m = fmaf(0.5f, v2m, hsum + b2h);   // v2 = 0.5 v2 + 0.5 h2
    bool s2 = v2m >= 1.0f;
    if (window) acc += s2 ? 1.0f : 0.0f;
    v2m = s2 ? 0.0f : v2m;
  };

  int t = 0;
  for (; t < dstart; t += 2) {           // outside decision window
    if (t + 64 < T)
      __builtin_prefetch(xb + (size_t)(t + 64) * 2, 0, 0);
    const float4 xv = *reinterpret_cast<const float4*>(xb + (size_t)t * 2);
    lif_step(xv.x, xv.y, false);
    lif_step(xv.z, xv.w, false);
  }
  for (; t < T; t += 2) {                // decision window: count s2 spikes
    const float4 xv = *reinterpret_cast<const float4*>(xb + (size_t)t * 2);
    lif_step(xv.x, xv.y, true);
    lif_step(xv.z, xv.w, true);
  }

  if (half == 0) out[b0 + m] = acc;      // O == 1
}

extern "C" void kernel_launch(void* const* d_in, const int* in_sizes, int n_in,
                              void* d_out, int out_size, void* d_ws, size_t ws_size,
                              hipStream_t stream) {
  const float* x  = (const float*)d_in[0];
  const float* W1 = (const float*)d_in[1];
  const float* b1 = (const float*)d_in[2];
  const float* W2 = (const float*)d_in[3];
  const float* b2 = (const float*)d_in[4];
  float* out = (float*)d_out;

  const int H = in_sizes[2];                         // 32
  const int O = in_sizes[4];                         // 1
  const int I = in_sizes[1] / H;                     // 2
  const int B = out_size / O;                        // 4096
  const int T = (int)((long long)in_sizes[0] / ((long long)B * I));  // 4096

  int dstart = T - T / 4;                            // max(T - T/4, T/2)
  if (T / 2 > dstart) dstart = T / 2;
  dstart &= ~1;                                      // even (true for T=4096)

  const int numTiles      = B / 16;                  // 256 waves total
  const int wavesPerBlock = 256 / 32;                // 8
  const int blocks        = (numTiles + wavesPerBlock - 1) / wavesPerBlock;

  snn_lif_wmma_kernel<<<blocks, 256, 0, stream>>>(
      x, W1, b1, W2, b2, out, T, dstart, numTiles);
}